// GRANMixtureBernoulli_2276332667422
// MI455X (gfx1250) — compile-verified
//
#include <hip/hip_runtime.h>

// GRAN mixture-Bernoulli loss for MI455X (gfx1250, wave32).
// Memory-bound segmented reduction: stream 703MB once (NT loads), scatter
// fp32 atomics into a 672KB L2-resident accumulator, tiny WMMA-assisted
// epilogue. fp32 throughout (log-prob magnitudes ~1e3 forbid fp16 paths).

#define S_SEG 4096   // NUM_SUBGRAPH (static in reference)
#define KCOMP 20     // mixture components

typedef __attribute__((ext_vector_type(2))) float v2f;
typedef __attribute__((ext_vector_type(4))) float v4f;
typedef __attribute__((ext_vector_type(8))) float v8f;

// ---------------- kernel 0: zero the accumulators ----------------
__global__ void zero_ws_kernel(float* __restrict__ ws, int n) {
    int i = blockIdx.x * blockDim.x + threadIdx.x;
    if (i < n) ws[i] = 0.0f;
}

// Numerically stable softplus: max(x,0) + log1p(exp(-|x|)).
// 1+t is exact enough here (t in (0,1]); lowers to v_exp_f32 / v_log_f32.
__device__ __forceinline__ float softplus_f(float x) {
    float t = __expf(-fabsf(x));
    return fmaxf(x, 0.0f) + __logf(1.0f + t);
}

// ---------------- kernel 1: streaming pass + scatter ----------------
// One edge per thread. Rows are 80B (20 fp32) and 16B aligned -> 5 x b128
// nontemporal loads per matrix. 41 fire-and-forget global_atomic_add_f32
// per edge into a 4096x41x4B = 672KB accumulator (L2 resident at 192MB).
__global__ void __launch_bounds__(256) gran_phase1_kernel(
    const float* __restrict__ label,
    const float* __restrict__ log_theta,
    const float* __restrict__ log_alpha,
    const int*   __restrict__ sidx,
    float* __restrict__ accA,   // [S][K]  sum of adj_loss
    float* __restrict__ accL,   // [S][K]  sum of log_alpha
    float* __restrict__ accC,   // [S]     edge counts
    int E)
{
    int stride = gridDim.x * blockDim.x;
    for (int e = blockIdx.x * blockDim.x + threadIdx.x; e < E; e += stride) {
        int   s = sidx[e];
        float y = label[e];
        const v4f* th = (const v4f*)(log_theta + (size_t)e * KCOMP);
        const v4f* la = (const v4f*)(log_alpha + (size_t)e * KCOMP);
        float* pa = accA + (size_t)s * KCOMP;
        float* pl = accL + (size_t)s * KCOMP;
#pragma unroll
        for (int q = 0; q < KCOMP / 4; ++q) {
            v4f t = __builtin_nontemporal_load(th + q);  // stream-once: NT hint
            v4f a = __builtin_nontemporal_load(la + q);
#pragma unroll
            for (int j = 0; j < 4; ++j) {
                float x   = t[j];
                float adj = softplus_f(x) - x * y;       // BCE-with-logits
                atomicAdd(pa + q * 4 + j, adj);
                atomicAdd(pl + q * 4 + j, a[j]);
            }
        }
        atomicAdd(accC + s, 1.0f);
    }
}

// ---------------- kernel 2: per-segment epilogue + WMMA reduction ----------
// Single block, 512 threads (16 waves). Each thread finishes 8 segments
// (divide, log_softmax over K, logsumexp over K). The 512 per-thread partial
// sums are then reduced by wave 0 with V_WMMA_F32_16X16X4_F32 vs an all-ones
// B matrix: each WMMA folds a 16x4 fp32 tile into replicated row sums,
// accumulating in the fp32 C matrix across 8 tiles (512 values total).
__global__ void __launch_bounds__(512) gran_phase2_kernel(
    const float* __restrict__ accA,
    const float* __restrict__ accL,
    const float* __restrict__ accC,
    float* __restrict__ out,
    float invE)
{
    __shared__ float red[512];
    int tid = threadIdx.x;

    float local = 0.0f;
    for (int s = tid; s < S_SEG; s += 512) {
        float c = accC[s];
        float x[KCOMP];
        float m1 = -3.402823466e38f;
#pragma unroll
        for (int k = 0; k < KCOMP; ++k) {
            x[k] = accL[(size_t)s * KCOMP + k] / c;   // mean log_alpha
            m1 = fmaxf(m1, x[k]);
        }
        float se = 0.0f;
#pragma unroll
        for (int k = 0; k < KCOMP; ++k) se += __expf(x[k] - m1);
        float lse1 = m1 + __logf(se);                 // log_softmax shift
        float m2 = -3.402823466e38f;
#pragma unroll
        for (int k = 0; k < KCOMP; ++k) {
            x[k] = (x[k] - lse1) - accA[(size_t)s * KCOMP + k];  // LS - adj
            m2 = fmaxf(m2, x[k]);
        }
        float s2 = 0.0f;
#pragma unroll
        for (int k = 0; k < KCOMP; ++k) s2 += __expf(x[k] - m2);
        local += m2 + __logf(s2);                     // log_prob[s]
    }
    red[tid] = local;
    __syncthreads();

    if (tid < 32) {   // whole wave 0 active -> EXEC all-ones for WMMA
        int lane = tid;
        int m    = lane & 15;            // A row this lane carries
        int koff = (lane >> 4) << 1;     // lanes 0-15: K=0,1; 16-31: K=2,3
        v2f b; b.x = 1.0f; b.y = 1.0f;   // B = ones(4x16)
        v8f d = {0.f, 0.f, 0.f, 0.f, 0.f, 0.f, 0.f, 0.f};
#pragma unroll
        for (int c = 0; c < 8; ++c) {    // 8 tiles x 64 values = 512 partials
            v2f a;
            a.x = red[c * 64 + m * 4 + koff + 0];
            a.y = red[c * 64 + m * 4 + koff + 1];
            // D[m][n] += rowsum_4(A[m]) for every n (B is all ones)
            d = __builtin_amdgcn_wmma_f32_16x16x4_f32(
                    /*neg_a=*/false, a, /*neg_b=*/false, b,
                    /*c_mod=*/(short)0, d, /*reuse_a=*/false, /*reuse_b=*/false);
        }
        // C/D layout: lanes 0-15 hold rows 0-7 in d[0..7], lanes 16-31 rows 8-15
        float rp = d[0] + d[1] + d[2] + d[3] + d[4] + d[5] + d[6] + d[7];
        float total = rp + __shfl_xor(rp, 16, 32);    // join the two halves
        if (lane == 0) out[0] = -total * invE;        // loss = -sum/E
    }
}

extern "C" void kernel_launch(void* const* d_in, const int* in_sizes, int n_in,
                              void* d_out, int out_size, void* d_ws, size_t ws_size,
                              hipStream_t stream) {
    const float* label     = (const float*)d_in[0];
    const float* log_theta = (const float*)d_in[1];
    const float* log_alpha = (const float*)d_in[2];
    const int*   sidx      = (const int*)d_in[3];
    int E = in_sizes[0];

    float* ws   = (float*)d_ws;
    float* accA = ws;                       // S*K floats
    float* accL = ws + (size_t)S_SEG * KCOMP;
    float* accC = ws + (size_t)2 * S_SEG * KCOMP;
    int nzero = S_SEG * (2 * KCOMP + 1);    // 672KB of accumulators

    zero_ws_kernel<<<(nzero + 255) / 256, 256, 0, stream>>>(ws, nzero);

    int blocks = (E + 255) / 256;
    if (blocks > 16384) blocks = 16384;
    gran_phase1_kernel<<<blocks, 256, 0, stream>>>(
        label, log_theta, log_alpha, sidx, accA, accL, accC, E);

    gran_phase2_kernel<<<1, 512, 0, stream>>>(
        accA, accL, accC, (float*)d_out, 1.0f / (float)E);
}